// SPNet_48704929137155
// MI455X (gfx1250) — compile-verified
//
#include <hip/hip_runtime.h>
#include <hip/hip_bf16.h>

// ---------------------------------------------------------------------------
// SPNet forward for MI455X (gfx1250, wave32).
// GEMM stages use V_WMMA_F32_16X16X4_F32. Per 64-wide K chunk: 16 A fragments
// are preloaded in one global clause (distinct regs -> loads overlap), then a
// dense ds_load_b64 + WMMA stream consumes them.
// Graph scatter uses global f32 atomics (HBM-bound part: ~1.4GB @ 23.3TB/s).
// ---------------------------------------------------------------------------

#define NEG_SLOPE 0.2f
#define NEG_INF_F (-1e9f)
#define SWS 160   // LDS stride (floats) per K-pair row; 160 % 64 == 32 -> the
                  // kk=0 half-wave and kk=2 half-wave hit disjoint bank halves

typedef __attribute__((ext_vector_type(2))) float v2f;
typedef __attribute__((ext_vector_type(8))) float v8f;

__device__ __forceinline__ float lrelu_f(float x) { return x > 0.f ? x : NEG_SLOPE * x; }

__device__ __forceinline__ void atomicMaxF(float* addr, float val) {
    unsigned int* ua = (unsigned int*)addr;
    unsigned int old = *((volatile unsigned int*)ua);
    while (__uint_as_float(old) < val) {
        unsigned int assumed = old;
        old = atomicCAS(ua, assumed, __float_as_uint(val));
        if (old == assumed) break;
    }
}

// One 64-wide K chunk. A fragments preloaded into 16 distinct v2f regs so the
// global loads issue as one clause and overlap the WMMA stream; B comes from
// pair-packed LDS. Must be force-inlined at call sites whose pointer
// expressions are rooted at kernarg pointers (A loads lower to GLOBAL).
__device__ __forceinline__ void gemm_chunk(const float* __restrict__ aptr,
                                           const float* __restrict__ sbase,
                                           v8f acc[4])
{
    v2f a[16];
    #pragma unroll
    for (int i = 0; i < 16; ++i)
        a[i] = *(const v2f*)(aptr + i * 4);        // K = {4i+kk, 4i+kk+1}
    #pragma unroll
    for (int i = 0; i < 16; ++i) {
        #pragma unroll
        for (int nt = 0; nt < 4; ++nt) {
            v2f b = *(const v2f*)(sbase + i * 2 * SWS + nt * 32);
            acc[nt] = __builtin_amdgcn_wmma_f32_16x16x4_f32(
                false, a[i], false, b, (short)0, acc[nt], false, false);
        }
    }
}

// ---------------------------------------------------------------------------
// Generic WMMA GEMM: C[M,64] = act(A[M,K] @ W[K,64] + bias)
//   - K is 64 or 128; weights row-major [K,64], staged pair-packed in LDS.
//   - Split-A mode: if A1 != nullptr, k<64 from A0[M,64], k>=64 from A1[M,64]
//     (implements concat(A0,A1) without materializing it).
//   - act: 0 = none, 1 = relu, 2 = leaky_relu(0.2)
// Block = 256 threads = 8 waves; each wave computes a 16x64 strip of C.
// OOB rows are handled by clamping the A row (their C rows are never stored),
// keeping EXEC full in the matrix loop. Chunk dispatch below is wave-uniform
// scalar branching, so EXEC stays all-ones through every WMMA.
// ---------------------------------------------------------------------------
__global__ __launch_bounds__(256)
void spnet_gemm(const float* __restrict__ A0, const float* __restrict__ A1,
                const float* __restrict__ W,  const float* __restrict__ bias,
                float* __restrict__ C, int M, int K, int act)
{
    __shared__ __align__(16) float sW2[64 * SWS];   // up to 64 K-pair rows (K=128)
    const int tid = threadIdx.x;
    // stage W pair-packed: sW2[p*SWS + n*2 + (k&1)] = W[k][n], p = k>>1
    for (int i = tid; i < K * 64; i += 256) {
        int k = i >> 6, n = i & 63;
        sW2[(k >> 1) * SWS + n * 2 + (k & 1)] = W[i];
    }
    __syncthreads();

    const int lane = tid & 31;
    const int wv   = tid >> 5;
    const int row0 = (blockIdx.x * 8 + wv) * 16;
    const int ml   = lane & 15;          // A row in tile / B,C col in tile
    const int kk   = (lane >> 4) << 1;   // K sub-offset 0 or 2 (VGPR-pair layout)

    // clamped A row: OOB rows compute garbage that is discarded at store
    const int rc = min(row0 + ml, M - 1);

    const float* sb0 = sW2 + (kk >> 1) * SWS + ml * 2;   // LDS base, chunk 0
    const float* sb1 = sb0 + 32 * SWS;                   // LDS base, chunk 1

    v8f acc[4] = {};   // 4 N-tiles of 16 -> 64 output columns

    // uniform scalar dispatch; every A pointer is rooted at a kernarg pointer
    if (A1 == nullptr) {
        if (K == 64) {
            gemm_chunk(A0 + (size_t)rc * 64 + kk, sb0, acc);
        } else {
            gemm_chunk(A0 + (size_t)rc * 128 + kk,      sb0, acc);
            gemm_chunk(A0 + (size_t)rc * 128 + 64 + kk, sb1, acc);
        }
    } else {
        gemm_chunk(A0 + (size_t)rc * 64 + kk, sb0, acc);
        gemm_chunk(A1 + (size_t)rc * 64 + kk, sb1, acc);
    }

    // C layout: VGPR j holds rows {j, j+8}; N = lane&15 within each 16-col tile
    const int mhi = (lane >> 4) << 3;
    #pragma unroll
    for (int j = 0; j < 8; ++j) {
        const int r = row0 + mhi + j;
        if (r < M) {
            #pragma unroll
            for (int nt = 0; nt < 4; ++nt) {
                const int n = nt * 16 + ml;
                float v = acc[nt][j];
                if (bias) v += bias[n];
                if (act == 1)      v = fmaxf(v, 0.f);
                else if (act == 2) v = lrelu_f(v);
                C[(size_t)r * 64 + n] = v;
            }
        }
    }
}

// ---------------------------------------------------------------------------
// Node-level helpers
// ---------------------------------------------------------------------------
__global__ void spnet_init_nodes(float* deg, float* emax, float* denom, int N) {
    int i = blockIdx.x * 256 + threadIdx.x;
    if (i < N) { deg[i] = 1.0f; emax[i] = NEG_INF_F; denom[i] = 0.0f; }
}

__global__ void spnet_zero(float* p, long long n) {
    long long i = (long long)blockIdx.x * 256 + threadIdx.x;
    if (i < n) p[i] = 0.0f;
}

__global__ void spnet_deg_count(const long long* __restrict__ dst, float* deg, int E) {
    int e = blockIdx.x * 256 + threadIdx.x;
    if (e < E) atomicAdd(&deg[dst[e]], 1.0f);
}

__global__ void spnet_dinv(const float* __restrict__ deg, float* dinv, int N) {
    int i = blockIdx.x * 256 + threadIdx.x;
    if (i < N) dinv[i] = 1.0f / sqrtf(fmaxf(deg[i], 1.0f));
}

// self-loop contribution of GCN conv: agg = dinv^2 * h  (both channels)
__global__ void spnet_gcn_self(const float* __restrict__ h_o, const float* __restrict__ h_t,
                               const float* __restrict__ dinv,
                               float* __restrict__ agg_o, float* __restrict__ agg_t, int N) {
    long long i = (long long)blockIdx.x * 256 + threadIdx.x;
    if (i < (long long)N * 64) {
        int node = (int)(i >> 6);
        float d = dinv[node];
        float w = d * d;
        agg_o[i] = w * h_o[i];
        agg_t[i] = w * h_t[i];
    }
}

// edge scatter of both GCN convs: one wave per edge, 2 features/lane/channel
__global__ void spnet_gcn_edge(const long long* __restrict__ src, const long long* __restrict__ dst,
                               const float* __restrict__ dinv,
                               const float* __restrict__ h_o, const float* __restrict__ h_t,
                               float* __restrict__ agg_o, float* __restrict__ agg_t, int E) {
    int gid  = blockIdx.x * 256 + threadIdx.x;
    int e    = gid >> 5;
    int lane = gid & 31;
    if (e >= E) return;
    long long s = src[e], d = dst[e];
    float w = dinv[s] * dinv[d];
    size_t sb = (size_t)s * 64, db = (size_t)d * 64;
    int f0 = lane, f1 = lane + 32;
    atomicAdd(&agg_o[db + f0], w * h_o[sb + f0]);
    atomicAdd(&agg_o[db + f1], w * h_o[sb + f1]);
    atomicAdd(&agg_t[db + f0], w * h_t[sb + f0]);
    atomicAdd(&agg_t[db + f1], w * h_t[sb + f1]);
}

// r = relu(agg + bias), in place, both channels
__global__ void spnet_relu_bias(float* agg_o, float* agg_t,
                                const float* __restrict__ b_go, const float* __restrict__ b_gt, int N) {
    long long i = (long long)blockIdx.x * 256 + threadIdx.x;
    if (i < (long long)N * 64) {
        int c = (int)(i & 63);
        agg_o[i] = fmaxf(agg_o[i] + b_go[c], 0.f);
        agg_t[i] = fmaxf(agg_t[i] + b_gt[c], 0.f);
    }
}

// s_sc / d_sc: dot of concat(r_o, r_t) with attention vectors
__global__ void spnet_attn_scores(const float* __restrict__ r_o, const float* __restrict__ r_t,
                                  const float* __restrict__ a_src, const float* __restrict__ a_dst,
                                  float* s_sc, float* d_sc, int N) {
    __shared__ float as[128], ad[128];
    int tid = threadIdx.x;
    if (tid < 128) { as[tid] = a_src[tid]; ad[tid] = a_dst[tid]; }
    __syncthreads();
    int i = blockIdx.x * 256 + tid;
    if (i >= N) return;
    const float* ro = r_o + (size_t)i * 64;
    const float* rt = r_t + (size_t)i * 64;
    float ss = 0.f, dd = 0.f;
    #pragma unroll 8
    for (int c = 0; c < 64; ++c) {
        ss += ro[c] * as[c] + rt[c] * as[64 + c];
        dd += ro[c] * ad[c] + rt[c] * ad[64 + c];
    }
    s_sc[i] = ss; d_sc[i] = dd;
}

// segment-max of masked edge scores (emax pre-initialized to NEG_INF)
__global__ void spnet_attn_max(const long long* __restrict__ src, const long long* __restrict__ dst,
                               const int* __restrict__ t,
                               const float* __restrict__ s_sc, const float* __restrict__ d_sc,
                               float* emax, int E) {
    int e = blockIdx.x * 256 + threadIdx.x;
    if (e >= E) return;
    long long s = src[e], d = dst[e];
    if (t[s] > 0) {
        float ee = lrelu_f(s_sc[s] + d_sc[d]);
        atomicMaxF(&emax[d], ee);
    }
}

// softmax numerator / denominator scatter: one wave per edge
__global__ void spnet_attn_accum(const long long* __restrict__ src, const long long* __restrict__ dst,
                                 const int* __restrict__ t,
                                 const float* __restrict__ s_sc, const float* __restrict__ d_sc,
                                 const float* __restrict__ emax, const float* __restrict__ r_t,
                                 float* denom, float* num, int E) {
    int gid  = blockIdx.x * 256 + threadIdx.x;
    int e    = gid >> 5;
    int lane = gid & 31;
    if (e >= E) return;
    long long s = src[e], d = dst[e];
    if (!(t[s] > 0)) return;                      // mask==0 contributes exactly 0
    float ee = lrelu_f(s_sc[s] + d_sc[d]);
    float w  = expf(ee - emax[d]);
    if (lane == 0) atomicAdd(&denom[d], w);
    size_t sb = (size_t)s * 64, db = (size_t)d * 64;
    int f0 = lane, f1 = lane + 32;
    atomicAdd(&num[db + f0], w * r_t[sb + f0]);
    atomicAdd(&num[db + f1], w * r_t[sb + f1]);
}

// h = num / (denom + 1e-16), in place on num
__global__ void spnet_attn_norm(float* num, const float* __restrict__ denom, int N) {
    long long i = (long long)blockIdx.x * 256 + threadIdx.x;
    if (i < (long long)N * 64) {
        int node = (int)(i >> 6);
        num[i] = num[i] / (denom[node] + 1e-16f);
    }
}

// final 64 -> 1 head:  out = (sig ? sigmoid : id)(in @ w + b)
__global__ void spnet_head(const float* __restrict__ in, const float* __restrict__ w,
                           const float* __restrict__ b, float* out, int N, int sig) {
    __shared__ float sw[64];
    int tid = threadIdx.x;
    if (tid < 64) sw[tid] = w[tid];
    __syncthreads();
    int i = blockIdx.x * 256 + tid;
    if (i >= N) return;
    const float* p = in + (size_t)i * 64;
    float acc = b[0];
    #pragma unroll 8
    for (int c = 0; c < 64; ++c) acc += p[c] * sw[c];
    if (sig) acc = 1.f / (1.f + expf(-acc));
    out[i] = acc;
}

__global__ void spnet_select(const int* __restrict__ t, const float* __restrict__ p1,
                             const float* __restrict__ p0, float* out, int N) {
    int i = blockIdx.x * 256 + threadIdx.x;
    if (i < N) out[i] = (t[i] > 0) ? p1[i] : p0[i];
}

// ---------------------------------------------------------------------------
// launcher
// ---------------------------------------------------------------------------
extern "C" void kernel_launch(void* const* d_in, const int* in_sizes, int n_in,
                              void* d_out, int out_size, void* d_ws, size_t ws_size,
                              hipStream_t stream) {
    const float*     x     = (const float*)d_in[0];
    const int*       t     = (const int*)d_in[1];
    const long long* ei    = (const long long*)d_in[3];
    const float* W_go = (const float*)d_in[4],  *b_go = (const float*)d_in[5];
    const float* W_gt = (const float*)d_in[6],  *b_gt = (const float*)d_in[7];
    const float* a_src = (const float*)d_in[8], *a_dst = (const float*)d_in[9];
    const float* W_enc = (const float*)d_in[10], *b_enc = (const float*)d_in[11];
    const float* d_w1 = (const float*)d_in[12], *d_b1 = (const float*)d_in[13];
    const float* d_w2 = (const float*)d_in[14], *d_b2 = (const float*)d_in[15];
    const float* d_w3 = (const float*)d_in[16], *d_b3 = (const float*)d_in[17];
    const float* p1_w1 = (const float*)d_in[18], *p1_b1 = (const float*)d_in[19];
    const float* p1_w2 = (const float*)d_in[20], *p1_b2 = (const float*)d_in[21];
    const float* p1_w3 = (const float*)d_in[22], *p1_b3 = (const float*)d_in[23];
    const float* p0_w1 = (const float*)d_in[24], *p0_b1 = (const float*)d_in[25];
    const float* p0_w2 = (const float*)d_in[26], *p0_b2 = (const float*)d_in[27];
    const float* p0_w3 = (const float*)d_in[28], *p0_b3 = (const float*)d_in[29];

    const int N = in_sizes[1];
    const int E = in_sizes[3] / 2;
    const long long* src = ei;
    const long long* dst = ei + E;

    // workspace layout (floats): 328*N total (~131 MB for N=100k)
    float* ws    = (float*)d_ws;
    float* deg   = ws;
    float* dinv  = ws + (size_t)1 * N;
    float* emax  = ws + (size_t)2 * N;
    float* denom = ws + (size_t)3 * N;
    float* s_sc  = ws + (size_t)4 * N;
    float* d_sc  = ws + (size_t)5 * N;
    float* pred1 = ws + (size_t)6 * N;
    float* pred0 = ws + (size_t)7 * N;
    float* h_o   = ws + (size_t)8 * N;    // 64N  (reused as MLP tmpA later)
    float* h_t   = ws + (size_t)72 * N;   // 64N  (reused as MLP tmpB later)
    float* agg_o = ws + (size_t)136 * N;  // 64N  -> becomes r_o in place
    float* agg_t = ws + (size_t)200 * N;  // 64N  -> becomes r_t in place
    float* num   = ws + (size_t)264 * N;  // 64N  -> becomes h (attention) in place

    float* out_pred_t = (float*)d_out;                  // [N]
    float* out_pred   = (float*)d_out + N;              // [N]
    float* reps       = (float*)d_out + (size_t)2 * N;  // [N,64]
    float* tmpA = h_o;
    float* tmpB = h_t;

    dim3 blk(256);
    int gN  = (N + 255) / 256;
    int gNF = (int)(((long long)N * 64 + 255) / 256);
    int gE  = (E + 255) / 256;
    int gEw = (E + 7) / 8;            // wave-per-edge, 8 waves per block
    int gM  = (N + 127) / 128;        // gemm: 128 rows per block

    // degree + normalization
    spnet_init_nodes<<<gN, blk, 0, stream>>>(deg, emax, denom, N);
    spnet_zero<<<gNF, blk, 0, stream>>>(num, (long long)N * 64);
    spnet_deg_count<<<gE, blk, 0, stream>>>(dst, deg, E);
    spnet_dinv<<<gN, blk, 0, stream>>>(deg, dinv, N);

    // h_o = x @ W_go ; h_t = x @ W_gt          (WMMA, K=128)
    spnet_gemm<<<gM, blk, 0, stream>>>(x, nullptr, W_go, nullptr, h_o, N, 128, 0);
    spnet_gemm<<<gM, blk, 0, stream>>>(x, nullptr, W_gt, nullptr, h_t, N, 128, 0);

    // GCN propagate (self-loop + edges), then r = relu(agg + b)
    spnet_gcn_self<<<gNF, blk, 0, stream>>>(h_o, h_t, dinv, agg_o, agg_t, N);
    spnet_gcn_edge<<<gEw, blk, 0, stream>>>(src, dst, dinv, h_o, h_t, agg_o, agg_t, E);
    spnet_relu_bias<<<gNF, blk, 0, stream>>>(agg_o, agg_t, b_go, b_gt, N);

    // masked attention
    spnet_attn_scores<<<gN, blk, 0, stream>>>(agg_o, agg_t, a_src, a_dst, s_sc, d_sc, N);
    spnet_attn_max<<<gE, blk, 0, stream>>>(src, dst, t, s_sc, d_sc, emax, E);
    spnet_attn_accum<<<gEw, blk, 0, stream>>>(src, dst, t, s_sc, d_sc, emax, agg_t, denom, num, E);
    spnet_attn_norm<<<gNF, blk, 0, stream>>>(num, denom, N);

    // reps = concat(r_o, h) @ W_enc + b_enc    (WMMA, split-A, K=128) -> d_out
    spnet_gemm<<<gM, blk, 0, stream>>>(agg_o, num, W_enc, b_enc, reps, N, 128, 0);

    // discriminator on r_t -> sigmoid -> pred_t
    spnet_gemm<<<gM, blk, 0, stream>>>(agg_t, nullptr, d_w1, d_b1, tmpA, N, 64, 2);
    spnet_gemm<<<gM, blk, 0, stream>>>(tmpA, nullptr, d_w2, d_b2, tmpB, N, 64, 2);
    spnet_head<<<gN, blk, 0, stream>>>(tmpB, d_w3, d_b3, out_pred_t, N, 1);

    // predictor 1 on reps
    spnet_gemm<<<gM, blk, 0, stream>>>(reps, nullptr, p1_w1, p1_b1, tmpA, N, 64, 2);
    spnet_gemm<<<gM, blk, 0, stream>>>(tmpA, nullptr, p1_w2, p1_b2, tmpB, N, 64, 2);
    spnet_head<<<gN, blk, 0, stream>>>(tmpB, p1_w3, p1_b3, pred1, N, 0);

    // predictor 0 on reps
    spnet_gemm<<<gM, blk, 0, stream>>>(reps, nullptr, p0_w1, p0_b1, tmpA, N, 64, 2);
    spnet_gemm<<<gM, blk, 0, stream>>>(tmpA, nullptr, p0_w2, p0_b2, tmpB, N, 64, 2);
    spnet_head<<<gN, blk, 0, stream>>>(tmpB, p0_w3, p0_b3, pred0, N, 0);

    spnet_select<<<gN, blk, 0, stream>>>(t, pred1, pred0, out_pred, N);
}